// GINELayer_30150670418203
// MI455X (gfx1250) — compile-verified
//
#include <hip/hip_runtime.h>
#include <math.h>

#define D 128
#define G 64
#define GN_EPS 1e-5f
#define NODES_PER_BLOCK 32

typedef float v2f __attribute__((ext_vector_type(2)));
typedef float v8f __attribute__((ext_vector_type(8)));
typedef int   v4i __attribute__((vector_size(4 * sizeof(int))));   // matches builtin param
typedef __attribute__((address_space(1))) v4i* as1_v4i_p;          // global
typedef __attribute__((address_space(3))) v4i* as3_v4i_p;          // LDS

#if __has_builtin(__builtin_amdgcn_global_load_async_to_lds_b128) && \
    __has_builtin(__builtin_amdgcn_s_wait_asynccnt)
#define ASYNC_STAGE 1
#else
#define ASYNC_STAGE 0
#endif

__device__ __forceinline__ float silu_f(float x) {
    return x / (1.0f + __expf(-x));
}

// Stage a 16-row x 128-col weight tile (rows tileN..tileN+15 of a [128,128]
// row-major matrix) into LDS with pitch 132 floats (bank-conflict-free
// float2 fragment reads; rows stay 16B-aligned: 132*4 = 33*16).
// Uses gfx1250 GLOBAL_LOAD_ASYNC_TO_LDS_B128 (ASYNCcnt) when available.
__device__ __forceinline__ void stage_b_tile(int tid, const float* __restrict__ w,
                                             int tileN, float* lds_b)
{
#if ASYNC_STAGE
    for (int idx = tid; idx < 16 * 32; idx += 256) {        // 512 x b128
        int j  = idx >> 5;                                  // row 0..15
        int k4 = (idx & 31) << 2;                           // col, step 4 floats
        __builtin_amdgcn_global_load_async_to_lds_b128(
            (as1_v4i_p)(w + (size_t)(tileN + j) * D + k4),
            (as3_v4i_p)(&lds_b[j * 132 + k4]),
            0, 0);
    }
    __builtin_amdgcn_s_wait_asynccnt(0);
#else
    for (int idx = tid; idx < 16 * 128; idx += 256) {
        int j = idx >> 7, k = idx & 127;
        lds_b[j * 132 + k] = w[(size_t)(tileN + j) * D + k];
    }
#endif
    __syncthreads();
}

// ---------------------------------------------------------------------------
// GEMM 1: h = node_h @ w1^T + b1        ([N,128] x [128,128]^T)
// One wave -> one 16x16 output tile via V_WMMA_F32_16X16X4_F32, K step 4.
// Block = 8 waves covering 128 rows x 16 cols.
// ---------------------------------------------------------------------------
__global__ __launch_bounds__(256) void lin_node_kernel(
    const float* __restrict__ node_h, const float* __restrict__ w1,
    const float* __restrict__ b1, float* __restrict__ h, int n_rows)
{
    __shared__ float lds_b[16 * 132];
    const int tid   = threadIdx.x;
    const int wave  = tid >> 5;
    const int lane  = tid & 31;
    const int tileN = blockIdx.y * 16;
    const int tileM = blockIdx.x * 128 + wave * 16;

    stage_b_tile(tid, w1, tileN, lds_b);

    const int lo16 = lane & 15;
    const int kOff = (lane >> 4) << 1;                  // 0 (lanes 0-15) / 2 (lanes 16-31)
    const int rowA = min(tileM + lo16, n_rows - 1);     // clamp keeps EXEC all-ones
    const float* aRow = node_h + (size_t)rowA * D;
    const float* bRow = &lds_b[lo16 * 132];

    v8f acc = {0.f, 0.f, 0.f, 0.f, 0.f, 0.f, 0.f, 0.f};
    #pragma unroll 8
    for (int k = 0; k < D; k += 4) {
        v2f a = *(const v2f*)(aRow + k + kOff);         // A frag: V0=K, V1=K+1 (hi half +2)
        v2f b = *(const v2f*)(bRow + k + kOff);         // B frag: same K striping
        acc = __builtin_amdgcn_wmma_f32_16x16x4_f32(
            false, a, false, b, (short)0, acc, false, false);
    }

    const int   col   = tileN + lo16;
    const int   rBase = tileM + ((lane >> 4) << 3);     // lanes 16-31 hold rows M+8..M+15
    const float bias  = b1[col];
    if (tileM + 16 <= n_rows) {                         // uniform fast path: no exec juggling
        #pragma unroll
        for (int r = 0; r < 8; ++r)
            h[(size_t)(rBase + r) * D + col] = acc[r] + bias;
    } else {
        #pragma unroll
        for (int r = 0; r < 8; ++r) {
            int row = rBase + r;
            if (row < n_rows) h[(size_t)row * D + col] = acc[r] + bias;
        }
    }
}

// ---------------------------------------------------------------------------
// GEMM 2 + message + scatter:
//   msg = SiLU(h[src] + edge_attr @ we^T + be);  agg[dst] += msg (atomic)
// ---------------------------------------------------------------------------
__global__ __launch_bounds__(256) void edge_msg_kernel(
    const float* __restrict__ edge_attr, const float* __restrict__ we,
    const float* __restrict__ be, const float* __restrict__ h,
    const int* __restrict__ src, const int* __restrict__ dst,
    float* __restrict__ agg, int n_edges)
{
    __shared__ float lds_b[16 * 132];
    const int tid   = threadIdx.x;
    const int wave  = tid >> 5;
    const int lane  = tid & 31;
    const int tileN = blockIdx.y * 16;
    const int tileM = blockIdx.x * 128 + wave * 16;

    stage_b_tile(tid, we, tileN, lds_b);

    const int lo16 = lane & 15;
    const int kOff = (lane >> 4) << 1;
    const int rowA = min(tileM + lo16, n_edges - 1);
    const float* aRow = edge_attr + (size_t)rowA * D;
    const float* bRow = &lds_b[lo16 * 132];

    v8f acc = {0.f, 0.f, 0.f, 0.f, 0.f, 0.f, 0.f, 0.f};
    #pragma unroll 8
    for (int k = 0; k < D; k += 4) {
        v2f a = *(const v2f*)(aRow + k + kOff);
        v2f b = *(const v2f*)(bRow + k + kOff);
        acc = __builtin_amdgcn_wmma_f32_16x16x4_f32(
            false, a, false, b, (short)0, acc, false, false);
    }

    const int   col   = tileN + lo16;
    const int   rBase = tileM + ((lane >> 4) << 3);
    const float bias  = be[col];
    if (tileM + 16 <= n_edges) {                        // uniform fast path
        #pragma unroll
        for (int r = 0; r < 8; ++r) {
            int e  = rBase + r;
            int s  = src[e];                            // uniform across 16-lane half
            int dd = dst[e];
            float v = silu_f(acc[r] + bias + h[(size_t)s * D + col]);
            atomicAdd(&agg[(size_t)dd * D + col], v);   // L2-resident scatter
        }
    } else {
        #pragma unroll
        for (int r = 0; r < 8; ++r) {
            int e = rBase + r;
            if (e < n_edges) {
                int s  = src[e];
                int dd = dst[e];
                float v = silu_f(acc[r] + bias + h[(size_t)s * D + col]);
                atomicAdd(&agg[(size_t)dd * D + col], v);
            }
        }
    }
}

// ---------------------------------------------------------------------------
// y = SiLU((1+eps)*h + agg) + node_h, plus per-graph feature sums and counts.
// batch is sorted, so each thread keeps a running sum over 32 nodes and
// flushes one atomic per graph transition -> ~30x fewer atomics.
// ---------------------------------------------------------------------------
__global__ __launch_bounds__(128) void fuse_y_kernel(
    const float* __restrict__ node_h, const float* __restrict__ h,
    const float* __restrict__ agg, const int* __restrict__ batch,
    const float* __restrict__ eps_p, float* __restrict__ y,
    float* __restrict__ sum, float* __restrict__ cnt, int n_rows)
{
    const int d  = threadIdx.x;
    const int n0 = blockIdx.x * NODES_PER_BLOCK;
    if (n0 >= n_rows) return;
    const int n1 = min(n0 + NODES_PER_BLOCK, n_rows);
    const float k1 = 1.0f + eps_p[0];

    int   curg = batch[n0];
    float run  = 0.f;
    float runc = 0.f;
    for (int n = n0; n < n1; ++n) {
        int g = batch[n];                          // uniform per iteration
        if (g != curg) {
            atomicAdd(&sum[curg * D + d], run);
            if (d == 0) atomicAdd(&cnt[curg], runc);
            run = 0.f; runc = 0.f; curg = g;
        }
        size_t off = (size_t)n * D + d;
        float conv = k1 * h[off] + agg[off];
        float yv   = silu_f(conv) + node_h[off];
        y[off] = yv;
        run  += yv;
        runc += 1.0f;
    }
    atomicAdd(&sum[curg * D + d], run);
    if (d == 0) atomicAdd(&cnt[curg], runc);
}

__global__ void mean_kernel(const float* __restrict__ sum,
                            const float* __restrict__ cnt,
                            float* __restrict__ mean)
{
    int i = blockIdx.x * blockDim.x + threadIdx.x;
    if (i < G * D) {
        float c = fmaxf(cnt[i >> 7], 1.0f);
        mean[i] = sum[i] / c;
    }
}

// t = y - mean[batch]*mean_scale (in place), plus per-graph sum of t^2.
__global__ __launch_bounds__(128) void center_kernel(
    float* __restrict__ y, const int* __restrict__ batch,
    const float* __restrict__ mean, const float* __restrict__ mean_scale,
    float* __restrict__ varsum, int n_rows)
{
    const int d  = threadIdx.x;
    const int n0 = blockIdx.x * NODES_PER_BLOCK;
    if (n0 >= n_rows) return;
    const int n1 = min(n0 + NODES_PER_BLOCK, n_rows);
    const float ms = mean_scale[d];

    int   curg = batch[n0];
    float m    = mean[curg * D + d];
    float run  = 0.f;
    for (int n = n0; n < n1; ++n) {
        int g = batch[n];
        if (g != curg) {
            atomicAdd(&varsum[curg * D + d], run);
            run = 0.f; curg = g; m = mean[g * D + d];
        }
        size_t off = (size_t)n * D + d;
        float t = y[off] - m * ms;
        y[off] = t;
        run += t * t;
    }
    atomicAdd(&varsum[curg * D + d], run);
}

__global__ void finalize_kernel(
    const float* __restrict__ t, const int* __restrict__ batch,
    const float* __restrict__ varsum, const float* __restrict__ cnt,
    const float* __restrict__ weight, const float* __restrict__ bias,
    float* __restrict__ out, int n_rows)
{
    size_t i = (size_t)blockIdx.x * blockDim.x + threadIdx.x;
    size_t total = (size_t)n_rows * D;
    if (i >= total) return;
    int n = (int)(i >> 7);
    int d = (int)(i & 127);
    int g = batch[n];
    float c   = fmaxf(cnt[g], 1.0f);
    float var = varsum[g * D + d] / c;
    float inv = 1.0f / sqrtf(var + GN_EPS);
    out[i] = weight[d] * t[i] * inv + bias[d];
}

// ---------------------------------------------------------------------------
extern "C" void kernel_launch(void* const* d_in, const int* in_sizes, int n_in,
                              void* d_out, int out_size, void* d_ws, size_t ws_size,
                              hipStream_t stream)
{
    const float* node_h    = (const float*)d_in[0];
    const float* edge_attr = (const float*)d_in[1];
    const int*   batch     = (const int*)d_in[2];
    const int*   ei        = (const int*)d_in[3];
    const float* w1        = (const float*)d_in[4];
    const float* b1        = (const float*)d_in[5];
    const float* we        = (const float*)d_in[6];
    const float* be        = (const float*)d_in[7];
    const float* eps       = (const float*)d_in[8];
    const float* gnw       = (const float*)d_in[9];
    const float* gnb       = (const float*)d_in[10];
    const float* gnms      = (const float*)d_in[11];

    const int N = in_sizes[0] / D;       // 50000
    const int E = in_sizes[1] / D;       // 600000
    const int* src = ei;                 // edge_index row 0
    const int* dst = ei + E;             // edge_index row 1

    // workspace layout (floats)
    float* ws     = (float*)d_ws;
    float* h      = ws;                          // N*D
    float* agg    = h   + (size_t)N * D;         // N*D
    float* y      = agg + (size_t)N * D;         // N*D (reused as t in place)
    float* sum    = y   + (size_t)N * D;         // G*D
    float* mean   = sum + G * D;                 // G*D
    float* varsum = mean + G * D;                // G*D
    float* cnt    = varsum + G * D;              // G

    (void)hipMemsetAsync(agg, 0, (size_t)N * D * sizeof(float), stream);
    (void)hipMemsetAsync(sum, 0, (size_t)(3 * G * D + G) * sizeof(float), stream);

    dim3 blk(256);
    dim3 g1((N + 127) / 128, D / 16);
    lin_node_kernel<<<g1, blk, 0, stream>>>(node_h, w1, b1, h, N);

    dim3 g2((E + 127) / 128, D / 16);
    edge_msg_kernel<<<g2, blk, 0, stream>>>(edge_attr, we, be, h, src, dst, agg, E);

    int nblocks = (N + NODES_PER_BLOCK - 1) / NODES_PER_BLOCK;
    fuse_y_kernel<<<nblocks, 128, 0, stream>>>(node_h, h, agg, batch, eps, y, sum, cnt, N);

    mean_kernel<<<(G * D + 255) / 256, 256, 0, stream>>>(sum, cnt, mean);

    center_kernel<<<nblocks, 128, 0, stream>>>(y, batch, mean, gnms, varsum, N);

    size_t total = (size_t)N * D;
    finalize_kernel<<<(int)((total + 255) / 256), 256, 0, stream>>>(
        y, batch, varsum, cnt, gnw, gnb, (float*)d_out, N);
}